// KAN_DiffPhys_ODE_23957327577256
// MI455X (gfx1250) — compile-verified
//
#include <hip/hip_runtime.h>
#include <math.h>

typedef float v2f __attribute__((ext_vector_type(2)));
typedef float v8f __attribute__((ext_vector_type(8)));

#define T_STEPS 1024
#define G1 50
#define G2 20

__device__ __forceinline__ float softplus_f(float x) {
    return (x > 20.0f) ? x : log1pf(__expf(x));
}

// ---------------------------------------------------------------------------
// Kernel 1: betas[t] = softplus(RBF_KAN2(RBF_KAN1(t)))  for t = 0..1023
// Layer-1 matmul  basis[1024x50(pad 52)] @ spline_w1[50x32]  done with
// V_WMMA_F32_16X16X4_F32 (fp32 in/out). One block = one 16-row tile of t,
// wave 0 -> output cols 0..15, wave 1 -> cols 16..31.
//
// WMMA f32 16x16x4 operand layouts (ISA 7.12.2):
//   A (16x4): lane l holds row M=l%16 (l&15); vgpr0 = K=(l<16?0:2), vgpr1 = K+1
//   B (4x16): vgpr v holds row K=v+(l<16?0:2), col N = l&15
//   C (16x16): vgpr v holds M=v+(l<16?0:8),  col N = l&15
// ---------------------------------------------------------------------------
__global__ void __launch_bounds__(64) kan_beta_kernel(
    const float* __restrict__ t_steps,    // [1024]
    const float* __restrict__ grid1,      // [50]
    const float* __restrict__ spline_w1,  // [50,32]
    const float* __restrict__ base_w1,    // [32]
    const float* __restrict__ grid2,      // [20]
    const float* __restrict__ spline_w2,  // [640]
    const float* __restrict__ base_w2,    // [32]
    float* __restrict__ betas)            // [1024] (workspace)
{
    __shared__ float h_lds[16 * 32];

    const int tid  = threadIdx.x;
    const int wave = tid >> 5;
    const int lane = tid & 31;
    const int half = lane >> 4;       // 0 or 1 (upper lane half)
    const int mn   = lane & 15;       // A row (M) / B-C column (N)
    const int t0   = blockIdx.x * 16;

    const float t_a  = t_steps[t0 + mn];  // t value for A-row M=mn
    const int   jcol = wave * 16 + mn;    // output column 0..31

    v8f c = {0.f, 0.f, 0.f, 0.f, 0.f, 0.f, 0.f, 0.f};

    // K = 50 padded to 52 -> 13 chunks of 4
    for (int kk = 0; kk < 13; ++kk) {
        const int k0 = kk * 4 + half * 2;
        const int k1 = k0 + 1;
        const float g0 = grid1[k0 < G1 ? k0 : 0];
        const float g1 = grid1[k1 < G1 ? k1 : 0];
        const float d0 = t_a - g0;
        const float d1 = t_a - g1;
        v2f a, b;
        a.x = (k0 < G1) ? __expf(-10.f * d0 * d0) : 0.f;
        a.y = (k1 < G1) ? __expf(-10.f * d1 * d1) : 0.f;
        b.x = (k0 < G1) ? spline_w1[k0 * 32 + jcol] : 0.f;
        b.y = (k1 < G1) ? spline_w1[k1 * 32 + jcol] : 0.f;
        // 8 args: (neg_a, A, neg_b, B, c_mod, C, reuse_a, reuse_b)
        c = __builtin_amdgcn_wmma_f32_16x16x4_f32(
                false, a, false, b, (short)0, c, false, false);
    }

    // h[t][j] = C + t * base_w1[j]; scatter this wave's 16 columns to LDS
    const float bw1 = base_w1[jcol];
#pragma unroll
    for (int v = 0; v < 8; ++v) {
        const int m = v + 8 * half;
        h_lds[m * 32 + jcol] = c[v] + t_steps[t0 + m] * bw1;
    }
    __syncthreads();

    // Layer 2: per-t reduction over j=0..31, g=0..19 (16 active lanes)
    if (tid < 16) {
        float acc = 0.f;
        for (int j = 0; j < 32; ++j) {
            const float hv = h_lds[tid * 32 + j];
            acc += hv * base_w2[j];
            const float* sw = spline_w2 + j * G2;
#pragma unroll
            for (int g = 0; g < G2; ++g) {
                const float d = hv - grid2[g];
                acc += __expf(-10.f * d * d) * sw[g];
            }
        }
        betas[t0 + tid] = softplus_f(acc);
    }
}

// ---------------------------------------------------------------------------
// Kernel 2: SIR Euler scan. One thread per batch element; betas cached in
// LDS (uniform broadcast reads); output streamed with non-temporal stores
// (write-once 128 MB -> keep it out of L2). Store-bandwidth bound.
// ---------------------------------------------------------------------------
__global__ void __launch_bounds__(256) sir_scan_kernel(
    const float* __restrict__ t_steps,
    const float* __restrict__ initial_I,
    const float* __restrict__ gamma_param,
    const float* __restrict__ betas,
    float* __restrict__ out,
    int Bsz)
{
    __shared__ float beta_lds[T_STEPS];
    const int tid = threadIdx.x;
#pragma unroll
    for (int i = 0; i < T_STEPS / 256; ++i)
        beta_lds[tid + i * 256] = betas[tid + i * 256];
    __syncthreads();

    const int b = blockIdx.x * 256 + tid;
    if (b >= Bsz) return;

    const float dt    = t_steps[1] - t_steps[0];
    const float gamma = softplus_f(gamma_param[0]);

    float I = initial_I[b];
    float S = 1.0f - I;

    float* outp = out + b;
    for (int t = 0; t < T_STEPS; ++t) {
        const float beta    = beta_lds[t];
        const float new_inf = beta * S * I;
        const float dI      = new_inf - gamma * I;
        I = fminf(fmaxf(fmaf(dt, dI, I), 0.f), 5.f);
        S = fminf(fmaxf(fmaf(-dt, new_inf, S), 0.f), 5.f);
        __builtin_nontemporal_store(I, outp);
        outp += Bsz;
    }
}

extern "C" void kernel_launch(void* const* d_in, const int* in_sizes, int n_in,
                              void* d_out, int out_size, void* d_ws, size_t ws_size,
                              hipStream_t stream) {
    const float* t_steps   = (const float*)d_in[0];
    const float* initial_I = (const float*)d_in[1];
    const float* grid1     = (const float*)d_in[2];
    const float* spline_w1 = (const float*)d_in[3];
    const float* base_w1   = (const float*)d_in[4];
    const float* grid2     = (const float*)d_in[5];
    const float* spline_w2 = (const float*)d_in[6];
    const float* base_w2   = (const float*)d_in[7];
    const float* gamma_p   = (const float*)d_in[8];

    float* betas = (float*)d_ws;   // 4 KB scratch
    float* out   = (float*)d_out;  // [1024, B]
    const int Bsz = in_sizes[1];

    kan_beta_kernel<<<T_STEPS / 16, 64, 0, stream>>>(
        t_steps, grid1, spline_w1, base_w1, grid2, spline_w2, base_w2, betas);

    sir_scan_kernel<<<(Bsz + 255) / 256, 256, 0, stream>>>(
        t_steps, initial_I, gamma_p, betas, out, Bsz);
}